// SignAdaptor_53017076302452
// MI455X (gfx1250) — compile-verified
//
#include <hip/hip_runtime.h>
#include <stdint.h>

// Reference shapes: B=32 videos, D_IMG=1024, D_CLIP=1024 -> 2048-wide output rows.
// Pure gather/concat/pad; bandwidth bound. Use gfx1250 async LDS copy path
// (global_load_async_to_lds_b128 / global_store_async_from_lds_b128, ASYNCcnt).

#define NB 32
#define DIMG 1024
#define DOUT 2048

// ---------------- setup: prefix sums of nf/nc + src_length tail ----------------
__global__ void va_setup_kernel(const int* __restrict__ nf, const int* __restrict__ nc,
                                int* __restrict__ offs, float* __restrict__ len_out) {
    __shared__ int s_nf[NB], s_nc[NB];
    const int t = threadIdx.x;           // 32 threads, 1 block (one wave32)
    s_nf[t] = nf[t];
    s_nc[t] = nc[t];
    __syncthreads();
    if (t == 0) {
        int fo = 0, co = 0;
        for (int i = 0; i < NB; ++i) {
            offs[i]      = fo;           // frame offset (exclusive cumsum of nf)
            offs[NB + i] = co;           // clip offset  (exclusive cumsum of nc)
            fo += s_nf[i];
            co += s_nc[i];
        }
    }
    if (len_out) len_out[t] = (float)s_nf[t];   // src_length tail of d_out
}

// ---------------- main gather: one output row (b, j) per 256-thread block ------
__global__ void va_gather_kernel(const float* __restrict__ img,
                                 const float* __restrict__ clip,
                                 const int*   __restrict__ nf,
                                 const int*   __restrict__ nc,
                                 const int*   __restrict__ offs,
                                 const int*   __restrict__ pad,
                                 float*       __restrict__ out,
                                 int L) {
    const int j = blockIdx.x;            // frame position 0..L-1
    const int b = blockIdx.y;            // video 0..31
    const int t = threadIdx.x;           // 0..255 ; each lane moves 2 x float4

    __shared__ float smem[DOUT];         // 8 KB staging (of 320 KB/WGP)

    float* outRow = out + ((size_t)b * (size_t)L + (size_t)j) * (size_t)DOUT;

    const int nfb = nf[b];               // uniform -> scalar loads
    if (j < nfb) {
        const int ncb  = nc[b];
        const int foff = offs[b];
        const int coff = offs[NB + b];
        const int rf   = nfb / ncb;      // repeat factor (>=1)
        int cidx = j / rf;
        const int cmax = ncb - 1;
        if (cidx > cmax) cidx = cmax;    // remainder tail clamps to last clip

        const float* imgRow  = img  + (size_t)(foff + j)    * (size_t)DIMG;
        const float* clipRow = clip + (size_t)(coff + cidx) * (size_t)DIMG;

        // LDS byte offsets for this lane's two 16B chunks
        unsigned lds_i = (unsigned)(uintptr_t)(&smem[t * 4]);
        unsigned lds_c = (unsigned)(uintptr_t)(&smem[DIMG + t * 4]);

        unsigned long long ga_i = (unsigned long long)(uintptr_t)(imgRow  + t * 4);
        unsigned long long ga_c = (unsigned long long)(uintptr_t)(clipRow + t * 4);

        // async global -> LDS (16B per lane, 512B per wave per instruction)
        asm volatile("global_load_async_to_lds_b128 %0, %1, off"
                     :: "v"(lds_i), "v"(ga_i) : "memory");
        asm volatile("global_load_async_to_lds_b128 %0, %1, off"
                     :: "v"(lds_c), "v"(ga_c) : "memory");
        // wait for both async loads of this wave to land in LDS
        asm volatile("s_wait_asynccnt 0" ::: "memory");

        unsigned long long go_i = (unsigned long long)(uintptr_t)(outRow + t * 4);
        unsigned long long go_c = (unsigned long long)(uintptr_t)(outRow + DIMG + t * 4);

        // async LDS -> global; data never touches VGPRs. S_ENDPGM waits idle.
        asm volatile("global_store_async_from_lds_b128 %0, %1, off"
                     :: "v"(go_i), "v"(lds_i) : "memory");
        asm volatile("global_store_async_from_lds_b128 %0, %1, off"
                     :: "v"(go_c), "v"(lds_c) : "memory");
    } else {
        // pad row: plain vector stores of pad value
        const float pv = (float)pad[0];
        float4 v = make_float4(pv, pv, pv, pv);
        float4* o4 = reinterpret_cast<float4*>(outRow);
        o4[t]       = v;                 // floats [0, 1024)
        o4[256 + t] = v;                 // floats [1024, 2048)
    }
}

extern "C" void kernel_launch(void* const* d_in, const int* in_sizes, int n_in,
                              void* d_out, int out_size, void* d_ws, size_t ws_size,
                              hipStream_t stream) {
    // setup_inputs order:
    //  0: image_batch [tf,1024] f32   1: emo_batch (UNUSED by reference)
    //  2: clip_batch  [tc,1024] f32   3: num_frames_batch [32] i32
    //  4: num_clips_batch [32] i32    5: pad_idx (scalar i32)
    const float* img  = (const float*)d_in[0];
    const float* clip = (const float*)d_in[2];
    const int*   nf   = (const int*)d_in[3];
    const int*   nc   = (const int*)d_in[4];
    const int*   pad  = (const int*)d_in[5];
    float*       out  = (float*)d_out;

    // Recover L from out_size: tuple output = 32*L*2048 floats + 32 lengths.
    long long os = (long long)out_size;
    long long rowset = (long long)NB * (long long)DOUT;   // 65536
    int L = 0;
    bool has_len = false;
    if (os > NB && ((os - NB) % rowset) == 0) {
        L = (int)((os - NB) / rowset);
        has_len = true;
    } else if (os > 0 && (os % rowset) == 0) {
        L = (int)(os / rowset);
    }
    if (L <= 0) return;

    int*   offs    = (int*)d_ws;                               // 64 ints scratch
    float* len_out = has_len ? (out + (size_t)NB * (size_t)L * (size_t)DOUT)
                             : nullptr;

    va_setup_kernel<<<1, NB, 0, stream>>>(nf, nc, offs, len_out);

    dim3 grid(L, NB);
    va_gather_kernel<<<grid, 256, 0, stream>>>(img, clip, nf, nc, offs, pad, out, L);
}